// TPSWarper_48773648613917
// MI455X (gfx1250) — compile-verified
//
#include <hip/hip_runtime.h>

// TPS warp, fused: WMMA einsum (offsets x RBF weights) + bilinear border gather.
// B=64, C=3, H=256, W=192, N_CTRL=25. f32 throughout (matches reference).

#define IMG_H   256
#define IMG_W   192
#define NCTRL   25
#define NBATCH  64
#define RBF_SC  10.0f
#define OFF_SC  0.3f

typedef __attribute__((ext_vector_type(2))) float v2f;
typedef __attribute__((ext_vector_type(8))) float v8f;

__global__ __launch_bounds__(128) void tps_warp_wmma_kernel(
    const float* __restrict__ cloth,   // (64,3,256,192)
    const float* __restrict__ theta,   // (64,50)
    float* __restrict__ out)           // (64,3,256,192)
{
    const int lane  = threadIdx.x & 31;
    const int wave  = threadIdx.x >> 5;
    const int tile  = blockIdx.x * 4 + wave;      // 16-pixel tile, 0..3071
    const int ncol  = lane & 15;                  // N (pixel within tile) / M (batch row) sublane
    const int khalf = (lane < 16) ? 0 : 2;        // which K sub-rows this lane holds

    // This wave's pixel (same for both half-waves): 16 contiguous x's on one row.
    const int pix = tile * 16 + ncol;
    const int py  = pix / IMG_W;
    const int px  = pix - py * IMG_W;
    const float mx = -1.0f + 2.0f * (float)px / (float)(IMG_W - 1);
    const float my = -1.0f + 2.0f * (float)py / (float)(IMG_H - 1);

    // ---- B matrix: RBF weights w[n, pix], K=25 padded to 28 (7 chunks of 4) ----
    v2f bw[7];
#pragma unroll
    for (int k = 0; k < 7; ++k) {
        float w0 = 0.0f, w1 = 0.0f;
        const int n0 = 4 * k + khalf;
        const int n1 = n0 + 1;
        if (n0 < NCTRL) {
            const float sx = -0.9f + 0.45f * (float)(n0 % 5);
            const float sy = -0.9f + 0.45f * (float)(n0 / 5);
            const float dx = mx - sx, dy = my - sy;
            w0 = __expf(-RBF_SC * (dx * dx + dy * dy));
        }
        if (n1 < NCTRL) {
            const float sx = -0.9f + 0.45f * (float)(n1 % 5);
            const float sy = -0.9f + 0.45f * (float)(n1 / 5);
            const float dx = mx - sx, dy = my - sy;
            w1 = __expf(-RBF_SC * (dx * dx + dy * dy));
        }
        bw[k].x = w0;
        bw[k].y = w1;
    }

    // ---- 4 batch-tiles of 16: D = A(offsets) x B(w), f32 WMMA 16x16x4 ----
#pragma unroll 1
    for (int bt = 0; bt < 4; ++bt) {
        const int abatch = bt * 16 + ncol;        // A-matrix row M for this lane
        v8f accx = {};
        v8f accy = {};
#pragma unroll
        for (int k = 0; k < 7; ++k) {
            const int n0 = 4 * k + khalf;
            const int n1 = n0 + 1;
            v2f t0 = {0.0f, 0.0f};
            v2f t1 = {0.0f, 0.0f};
            if (n0 < NCTRL) t0 = *(const v2f*)(theta + abatch * (2 * NCTRL) + 2 * n0);
            if (n1 < NCTRL) t1 = *(const v2f*)(theta + abatch * (2 * NCTRL) + 2 * n1);
            v2f ax = { t0.x * OFF_SC, t1.x * OFF_SC };
            v2f ay = { t0.y * OFF_SC, t1.y * OFF_SC };
            accx = __builtin_amdgcn_wmma_f32_16x16x4_f32(
                false, ax, false, bw[k], (short)0, accx, false, false);
            accy = __builtin_amdgcn_wmma_f32_16x16x4_f32(
                false, ay, false, bw[k], (short)0, accy, false, false);
        }

        // ---- bilinear border-clamped gather for this 16x16 (batch x pixel) tile ----
#pragma unroll
        for (int v = 0; v < 8; ++v) {
            const int sb = bt * 16 + v + ((lane < 16) ? 0 : 8);   // D-row -> batch
            float gx = fminf(fmaxf(mx + accx[v], -1.0f), 1.0f);
            float gy = fminf(fmaxf(my + accy[v], -1.0f), 1.0f);
            const float xf = (gx + 1.0f) * 0.5f * (float)(IMG_W - 1);
            const float yf = (gy + 1.0f) * 0.5f * (float)(IMG_H - 1);
            const float x0f = floorf(xf);
            const float y0f = floorf(yf);
            const float wx = xf - x0f;
            const float wy = yf - y0f;
            int x0 = (int)x0f;
            int y0 = (int)y0f;
            const int x1 = min(max(x0 + 1, 0), IMG_W - 1);
            const int y1 = min(max(y0 + 1, 0), IMG_H - 1);
            x0 = min(max(x0, 0), IMG_W - 1);
            y0 = min(max(y0, 0), IMG_H - 1);
            const float wa = (1.0f - wx) * (1.0f - wy);
            const float wb = wx * (1.0f - wy);
            const float wc = (1.0f - wx) * wy;
            const float wd = wx * wy;
#pragma unroll
            for (int c = 0; c < 3; ++c) {
                const float* bp = cloth + (size_t)(sb * 3 + c) * (IMG_H * IMG_W);
                const float Ia = bp[y0 * IMG_W + x0];
                const float Ib = bp[y0 * IMG_W + x1];
                const float Ic = bp[y1 * IMG_W + x0];
                const float Id = bp[y1 * IMG_W + x1];
                out[(size_t)(sb * 3 + c) * (IMG_H * IMG_W) + pix] =
                    wa * Ia + wb * Ib + wc * Ic + wd * Id;
            }
        }
    }
}

extern "C" void kernel_launch(void* const* d_in, const int* in_sizes, int n_in,
                              void* d_out, int out_size, void* d_ws, size_t ws_size,
                              hipStream_t stream) {
    (void)in_sizes; (void)n_in; (void)d_ws; (void)ws_size; (void)out_size;
    const float* cloth = (const float*)d_in[0];
    const float* theta = (const float*)d_in[1];
    float* out = (float*)d_out;

    // 49152 pixels / 16 per wave = 3072 waves; 4 waves (128 threads) per block.
    dim3 grid(768), block(128);
    tps_warp_wmma_kernel<<<grid, block, 0, stream>>>(cloth, theta, out);
}